// NodeModel_1_23630910063281
// MI455X (gfx1250) — compile-verified
//
#include <hip/hip_runtime.h>

typedef float v2f __attribute__((ext_vector_type(2)));
typedef float v8f __attribute__((ext_vector_type(8)));

#define NN 50000
#define NE 800000

// workspace layout (float offsets)
#define SUM1_OFF  0        // [128] sum h
#define SUM2_OFF  128      // [128] sum h^2
#define SCALE_OFF 256      // [128]
#define SHIFT_OFF 384      // [128]
#define CNT_OFF   512      // [NN]
#define AGG_OFF   51200    // [NN*128]
#define WS_FLOATS (AGG_OFF + NN * 128)

__device__ __forceinline__ v8f wmma4(v2f a, v2f b, v8f c) {
  return __builtin_amdgcn_wmma_f32_16x16x4_f32(false, a, false, b, (short)0, c,
                                               false, false);
}

// ---------------- k0: zero workspace ----------------
__global__ void k0_zero(float* __restrict__ p, int n) {
  int i = blockIdx.x * blockDim.x + threadIdx.x;
  int stride = gridDim.x * blockDim.x;
  for (; i < n; i += stride) p[i] = 0.0f;
}

// ---------------- k1: edge GEMM + BN partial sums ----------------
// block = 256 threads = 8 waves; wave = 16 edges x 128 cols
__global__ __launch_bounds__(256) void k1_edge_stats(
    const float* __restrict__ x, const int* __restrict__ eidx,
    const float* __restrict__ ea, const float* __restrict__ W1,
    const float* __restrict__ b1, float* __restrict__ ws) {
  __shared__ float sW1p[44 * 128 * 2];   // K-pair packed W1
  __shared__ float sFeat[8 * 16 * 92];   // per-wave 16x88 feature tiles
  __shared__ int   sCol[128];
  __shared__ float sSums[256];

  const int tid = threadIdx.x;
  const int w = tid >> 5, lane = tid & 31;
  const int laneM = lane & 15, hi = lane >> 4;

  // stage W1, packed as pairs along K: sW1p[((k>>1)*128 + c)*2 + (k&1)]
  for (int i = tid; i < 88 * 128; i += 256) {
    int k = i >> 7, c = i & 127;
    sW1p[(((k >> 1) * 128) + c) * 2 + (k & 1)] = W1[i];
  }
  sSums[tid] = 0.0f;

  const int e0 = blockIdx.x * 128 + w * 16;  // NE % 128 == 0: no tail
  if (lane < 16) sCol[w * 16 + lane] = eidx[NE + e0 + lane];

  // stage per-edge features [16 x 88] (wave-private region)
  for (int i = lane; i < 16 * 88; i += 32) {
    int m = i / 88, k = i - m * 88;
    float v = (k < 4) ? x[sCol[w * 16 + m] * 4 + k]
                      : ea[(e0 + m) * 84 + (k - 4)];
    sFeat[(w * 16 + m) * 92 + k] = v;
  }
  __syncthreads();

  v8f acc[8];
  const v8f vz = {0.f, 0.f, 0.f, 0.f, 0.f, 0.f, 0.f, 0.f};
#pragma unroll
  for (int nt = 0; nt < 8; nt++) acc[nt] = vz;

  const float* fb = sFeat + (w * 16 + laneM) * 92;
  const int koff = hi * 2;
  const v2f* w1p = (const v2f*)sW1p;
#pragma unroll
  for (int k0 = 0; k0 < 88; k0 += 4) {
    v2f a = *(const v2f*)(fb + k0 + koff);  // 8B aligned (even offset)
    int kk2 = (k0 + koff) >> 1;
#pragma unroll
    for (int nt = 0; nt < 8; nt++) {
      v2f b = w1p[kk2 * 128 + nt * 16 + laneM];
      acc[nt] = wmma4(a, b, acc[nt]);
    }
  }

  // epilogue: add bias, reduce sum / sum-of-squares per column
#pragma unroll
  for (int nt = 0; nt < 8; nt++) {
    float bv = b1[nt * 16 + laneM];
    float s1 = 0.f, s2 = 0.f;
#pragma unroll
    for (int r = 0; r < 8; r++) {
      float v = acc[nt][r] + bv;
      s1 += v;
      s2 += v * v;
    }
    s1 += __shfl_xor(s1, 16, 32);  // lane L and L^16 share column N = L&15
    s2 += __shfl_xor(s2, 16, 32);
    if (lane < 16) {
      unsafeAtomicAdd(&sSums[nt * 16 + lane], s1);
      unsafeAtomicAdd(&sSums[128 + nt * 16 + lane], s2);
    }
  }
  __syncthreads();
  unsafeAtomicAdd(&ws[SUM1_OFF + tid], sSums[tid]);  // tid 0..255 covers S1,S2
}

// ---------------- k2: finalize BN affine ----------------
__global__ void k2_finalize(const float* __restrict__ gamma,
                            const float* __restrict__ beta,
                            float* __restrict__ ws) {
  int c = threadIdx.x;
  if (c < 128) {
    const float invE = 1.0f / (float)NE;
    float m = ws[SUM1_OFF + c] * invE;
    float var = ws[SUM2_OFF + c] * invE - m * m;
    float inv = rsqrtf(var + 1e-5f);
    float sc = gamma[c] * inv;
    ws[SCALE_OFF + c] = sc;
    ws[SHIFT_OFF + c] = beta[c] - m * sc;
  }
}

// ---------------- k3: edge GEMM (recompute) + BN + scatter ----------------
__global__ __launch_bounds__(256) void k3_edge_scatter(
    const float* __restrict__ x, const int* __restrict__ eidx,
    const float* __restrict__ ea, const float* __restrict__ W1,
    const float* __restrict__ b1, float* __restrict__ ws) {
  __shared__ float sW1p[44 * 128 * 2];
  __shared__ float sFeat[8 * 16 * 92];
  __shared__ int   sCol[128];
  __shared__ int   sRow[128];

  const int tid = threadIdx.x;
  const int w = tid >> 5, lane = tid & 31;
  const int laneM = lane & 15, hi = lane >> 4;

  for (int i = tid; i < 88 * 128; i += 256) {
    int k = i >> 7, c = i & 127;
    sW1p[(((k >> 1) * 128) + c) * 2 + (k & 1)] = W1[i];
  }

  const int e0 = blockIdx.x * 128 + w * 16;
  if (lane < 16) {
    int rw = eidx[e0 + lane];
    sRow[w * 16 + lane] = rw;
    sCol[w * 16 + lane] = eidx[NE + e0 + lane];
    unsafeAtomicAdd(&ws[CNT_OFF + rw], 1.0f);  // one count per edge
  }
  for (int i = lane; i < 16 * 88; i += 32) {
    int m = i / 88, k = i - m * 88;
    float v = (k < 4) ? x[sCol[w * 16 + m] * 4 + k]
                      : ea[(e0 + m) * 84 + (k - 4)];
    sFeat[(w * 16 + m) * 92 + k] = v;
  }
  __syncthreads();

  v8f acc[8];
  const v8f vz = {0.f, 0.f, 0.f, 0.f, 0.f, 0.f, 0.f, 0.f};
#pragma unroll
  for (int nt = 0; nt < 8; nt++) acc[nt] = vz;

  const float* fb = sFeat + (w * 16 + laneM) * 92;
  const int koff = hi * 2;
  const v2f* w1p = (const v2f*)sW1p;
#pragma unroll
  for (int k0 = 0; k0 < 88; k0 += 4) {
    v2f a = *(const v2f*)(fb + k0 + koff);
    int kk2 = (k0 + koff) >> 1;
#pragma unroll
    for (int nt = 0; nt < 8; nt++) {
      v2f b = w1p[kk2 * 128 + nt * 16 + laneM];
      acc[nt] = wmma4(a, b, acc[nt]);
    }
  }

  int rowm[8];
#pragma unroll
  for (int r = 0; r < 8; r++) rowm[r] = sRow[w * 16 + r + hi * 8];

  float* agg = ws + AGG_OFF;
#pragma unroll
  for (int nt = 0; nt < 8; nt++) {
    int c = nt * 16 + laneM;
    float bv = b1[c];
    float sc = ws[SCALE_OFF + c];
    float sh = ws[SHIFT_OFF + c];
#pragma unroll
    for (int r = 0; r < 8; r++) {
      float v = (acc[nt][r] + bv) * sc + sh;
      unsafeAtomicAdd(agg + (long)rowm[r] * 128 + c, v);
    }
  }
}

// ---------------- k4: node GEMM [N,132]@[132,64] + ReLU ----------------
__global__ __launch_bounds__(256) void k4_node_gemm(
    const float* __restrict__ x, const float* __restrict__ W2,
    const float* __restrict__ b2, const float* __restrict__ ws,
    float* __restrict__ out) {
  __shared__ float sW2p[66 * 64 * 2];    // K-pair packed W2 (132x64)
  __shared__ float sFeat[8 * 16 * 137];  // 16x132 node feature tiles
  __shared__ float sInv[128];

  const int tid = threadIdx.x;
  const int w = tid >> 5, lane = tid & 31;
  const int laneM = lane & 15, hi = lane >> 4;

  for (int i = tid; i < 132 * 64; i += 256) {
    int k = i >> 6, c = i & 63;
    sW2p[(((k >> 1) * 64) + c) * 2 + (k & 1)] = W2[i];
  }

  const int n0 = blockIdx.x * 128 + w * 16;
  const float* cnt = ws + CNT_OFF;
  const float* agg = ws + AGG_OFF;
  if (lane < 16) {
    int n = n0 + lane;
    float c = (n < NN) ? cnt[n] : 1.0f;
    sInv[w * 16 + lane] = 1.0f / fmaxf(c, 1.0f);
  }
  for (int i = lane; i < 16 * 132; i += 32) {
    int m = i / 132, k = i - m * 132;
    int n = n0 + m;
    float v = 0.0f;
    if (n < NN)
      v = (k < 4) ? x[n * 4 + k] : agg[(long)n * 128 + (k - 4)] * sInv[w * 16 + m];
    sFeat[(w * 16 + m) * 137 + k] = v;
  }
  __syncthreads();

  v8f acc[4];
  const v8f vz = {0.f, 0.f, 0.f, 0.f, 0.f, 0.f, 0.f, 0.f};
#pragma unroll
  for (int nt = 0; nt < 4; nt++) acc[nt] = vz;

  const float* fb = sFeat + (w * 16 + laneM) * 137;
  const int koff = hi * 2;
  const v2f* w2p = (const v2f*)sW2p;
#pragma unroll
  for (int k0 = 0; k0 < 132; k0 += 4) {
    v2f a;
    a.x = fb[k0 + koff];      // odd stride -> scalar LDS loads
    a.y = fb[k0 + koff + 1];
    int kk2 = (k0 + koff) >> 1;
#pragma unroll
    for (int nt = 0; nt < 4; nt++) {
      v2f b = w2p[kk2 * 64 + nt * 16 + laneM];
      acc[nt] = wmma4(a, b, acc[nt]);
    }
  }

#pragma unroll
  for (int nt = 0; nt < 4; nt++) {
    int c = nt * 16 + laneM;
    float bv = b2[c];
#pragma unroll
    for (int r = 0; r < 8; r++) {
      int n = n0 + r + hi * 8;
      if (n < NN) out[n * 64 + c] = fmaxf(acc[nt][r] + bv, 0.0f);
    }
  }
}

extern "C" void kernel_launch(void* const* d_in, const int* in_sizes, int n_in,
                              void* d_out, int out_size, void* d_ws,
                              size_t ws_size, hipStream_t stream) {
  const float* x     = (const float*)d_in[0];
  const int*   eidx  = (const int*)d_in[1];
  const float* ea    = (const float*)d_in[2];
  const float* W1    = (const float*)d_in[3];
  const float* b1    = (const float*)d_in[4];
  const float* gamma = (const float*)d_in[5];
  const float* beta  = (const float*)d_in[6];
  const float* W2    = (const float*)d_in[7];
  const float* b2    = (const float*)d_in[8];
  float* ws  = (float*)d_ws;
  float* out = (float*)d_out;

  k0_zero<<<2048, 256, 0, stream>>>(ws, WS_FLOATS);
  k1_edge_stats<<<NE / 128, 256, 0, stream>>>(x, eidx, ea, W1, b1, ws);
  k2_finalize<<<1, 128, 0, stream>>>(gamma, beta, ws);
  k3_edge_scatter<<<NE / 128, 256, 0, stream>>>(x, eidx, ea, W1, b1, ws);
  k4_node_gemm<<<(NN + 127) / 128, 256, 0, stream>>>(x, W2, b2, ws, out);
}